// VectorQuantizer_66889820668041
// MI455X (gfx1250) — compile-verified
//
#include <hip/hip_runtime.h>
#include <hip/hip_bf16.h>
#include <cstdint>

typedef __attribute__((ext_vector_type(2))) float v2f;
typedef __attribute__((ext_vector_type(8))) float v8f;

#define NUM_CODES 8192
#define CODE_DIM  512
#define NROWS     32768              // 8*4096
#define K_TILES   (NUM_CODES / 16)   // 512
#define WAVES_PB  8
#define TILES_PW  (K_TILES / WAVES_PB) // 64 code tiles per wave
#define NT        4                  // code tiles processed together
#define GROUPS    (TILES_PW / NT)    // 16
#define MT        2                  // 16-row tiles per block (32 rows)
#define ROW_PITCH 514                // LDS row pitch (floats), avoids bank conflicts

// ---------------- Kernel 0: c_sq[k] = sum_d codebook[k][d]^2 ----------------
__global__ void vq_csq_kernel(const float* __restrict__ cb, float* __restrict__ csq) {
    const int k = blockIdx.x;
    const int t = threadIdx.x; // 128 threads
    const float* row = cb + (size_t)k * CODE_DIM;
    float a0 = row[t], a1 = row[t + 128], a2 = row[t + 256], a3 = row[t + 384];
    float s = (a0 * a0 + a1 * a1) + (a2 * a2 + a3 * a3);
    __shared__ float red[128];
    red[t] = s;
    __syncthreads();
    for (int o = 64; o > 0; o >>= 1) {
        if (t < o) red[t] += red[t + o];
        __syncthreads();
    }
    if (t == 0) csq[k] = red[0];
}

// ---------------- Kernel 1: WMMA distance matmul + argmin ----------------
__global__ __launch_bounds__(256) void vq_argmin_kernel(
    const float* __restrict__ ze, const float* __restrict__ cb,
    const float* __restrict__ csq, float* __restrict__ outIdx) {

    __shared__ float sA[32 * ROW_PITCH];   // 32 rows of z_e, padded pitch
    __shared__ float sZsq[32];
    __shared__ float sPart[256];
    __shared__ float sRedV[WAVES_PB][32];
    __shared__ int   sRedI[WAVES_PB][32];

    const int tid = threadIdx.x;
    const int m0  = blockIdx.x * 32;

    // ---- stage 32x512 z_e tile into LDS (coalesced float4) ----
    {
        const float4* src = (const float4*)(ze + (size_t)m0 * CODE_DIM);
        #pragma unroll
        for (int it = 0; it < 16; ++it) {
            int i = tid + it * 256;          // 4096 float4 total
            float4 v = src[i];
            int e   = i * 4;
            int row = e >> 9;                // /512
            int col = e & 511;
            float* dst = &sA[row * ROW_PITCH + col];
            dst[0] = v.x; dst[1] = v.y; dst[2] = v.z; dst[3] = v.w;
        }
    }
    __syncthreads();

    // ---- z_sq per row, tree-style (mimic XLA reduction) ----
    {
        int row = tid >> 3;                  // 32 rows
        int seg = tid & 7;                   // 8 segments of 64 elems
        const float* p = &sA[row * ROW_PITCH + seg * 64];
        float l0[32];
        #pragma unroll
        for (int i = 0; i < 32; ++i) {
            float a = p[2 * i], b = p[2 * i + 1];
            l0[i] = a * a + b * b;
        }
        #pragma unroll
        for (int st = 16; st > 0; st >>= 1)
            #pragma unroll
            for (int i = 0; i < 16; ++i)
                if (i < st) l0[i] += l0[i + st];
        sPart[row * 8 + seg] = l0[0];
    }
    __syncthreads();
    if (tid < 32) {
        const float* p = &sPart[tid * 8];
        sZsq[tid] = ((p[0] + p[1]) + (p[2] + p[3])) + ((p[4] + p[5]) + (p[6] + p[7]));
    }
    __syncthreads();

    const int lane = tid & 31;
    const int wave = tid >> 5;
    const int half = lane >> 4;
    const int lp   = lane & 15;

    // A fragment LDS offsets for the two 16-row sub-tiles
    const int aoff0 = (lp)      * ROW_PITCH + 2 * half;
    const int aoff1 = (16 + lp) * ROW_PITCH + 2 * half;

    float bestv[MT][8];
    int   besti[MT][8];
    #pragma unroll
    for (int mt = 0; mt < MT; ++mt)
        #pragma unroll
        for (int j = 0; j < 8; ++j) { bestv[mt][j] = 3.4e38f; besti[mt][j] = 0; }

    for (int g = 0; g < GROUPS; ++g) {
        const int tbase = wave * TILES_PW + g * NT;
        const float* cbp[NT];
        float csql[NT];
        int   n0[NT];
        #pragma unroll
        for (int nt = 0; nt < NT; ++nt) {
            n0[nt]   = (tbase + nt) * 16;
            cbp[nt]  = cb + (size_t)(n0[nt] + lp) * CODE_DIM + 2 * half;
            csql[nt] = csq[n0[nt] + lp];
        }

        v8f acc[NT][MT];
        #pragma unroll
        for (int nt = 0; nt < NT; ++nt)
            #pragma unroll
            for (int mt = 0; mt < MT; ++mt)
                acc[nt][mt] = 0;

        #pragma unroll 4
        for (int d = 0; d < CODE_DIM; d += 4) {
            v2f a0 = *(const v2f*)(&sA[aoff0 + d]);
            v2f a1 = *(const v2f*)(&sA[aoff1 + d]);
            #pragma unroll
            for (int nt = 0; nt < NT; ++nt) {
                v2f b = *(const v2f*)(cbp[nt] + d);
                acc[nt][0] = __builtin_amdgcn_wmma_f32_16x16x4_f32(
                    false, a0, false, b, (short)0, acc[nt][0], false, false);
                acc[nt][1] = __builtin_amdgcn_wmma_f32_16x16x4_f32(
                    false, a1, false, b, (short)0, acc[nt][1], false, false);
            }
        }

        // distances: (z_sq - 2*dot) + c_sq, same association as reference
        #pragma unroll
        for (int nt = 0; nt < NT; ++nt) {
            const int idx = n0[nt] + lp;
            #pragma unroll
            for (int mt = 0; mt < MT; ++mt) {
                #pragma unroll
                for (int j = 0; j < 8; ++j) {
                    const int m = j + 8 * half + 16 * mt;
                    float v = (sZsq[m] - 2.0f * acc[nt][mt][j]) + csql[nt];
                    if (v < bestv[mt][j] ||
                        (v == bestv[mt][j] && idx < besti[mt][j])) {
                        bestv[mt][j] = v; besti[mt][j] = idx;
                    }
                }
            }
        }
    }

    // ---- cross-lane argmin within 16-lane halves (C layout: N = lane%16) ----
    #pragma unroll
    for (int mt = 0; mt < MT; ++mt) {
        #pragma unroll
        for (int j = 0; j < 8; ++j) {
            float v = bestv[mt][j];
            int idx = besti[mt][j];
            #pragma unroll
            for (int msk = 8; msk >= 1; msk >>= 1) {
                float ov = __shfl_xor(v, msk, 32);
                int   oi = __shfl_xor(idx, msk, 32);
                if (ov < v || (ov == v && oi < idx)) { v = ov; idx = oi; }
            }
            if (lp == 0) {
                const int m = j + 8 * half + 16 * mt;
                sRedV[wave][m] = v;
                sRedI[wave][m] = idx;
            }
        }
    }
    __syncthreads();

    // ---- cross-wave argmin, write indices ----
    if (tid < 32) {
        float v = sRedV[0][tid];
        int idx  = sRedI[0][tid];
        #pragma unroll
        for (int w = 1; w < WAVES_PB; ++w) {
            float ov = sRedV[w][tid];
            int   oi = sRedI[w][tid];
            if (ov < v || (ov == v && oi < idx)) { v = ov; idx = oi; }
        }
        outIdx[m0 + tid] = (float)idx;   // exact for idx < 2^24
    }
}

// ---------------- Kernel 2: gather z_q, write z_q_st, per-row loss partial ----------------
__global__ void vq_gather_kernel(const float* __restrict__ ze,
                                 const float* __restrict__ cb,
                                 const float* __restrict__ idxF,
                                 float* __restrict__ zq_out,
                                 float* __restrict__ partial) {
    const int row = blockIdx.x;
    const int t   = threadIdx.x; // 128
    const int idx = (int)idxF[row];
    const float* crow = cb + (size_t)idx * CODE_DIM;
    const float* zrow = ze + (size_t)row * CODE_DIM;
    float* orow = zq_out + (size_t)row * CODE_DIM;
    float a[4];
    #pragma unroll
    for (int i = 0; i < 4; ++i) {
        int c = t + i * 128;
        float q = crow[c], z = zrow[c];
        orow[c] = z + (q - z);           // match reference straight-through op
        float d = q - z;
        a[i] = d * d;
    }
    float s = (a[0] + a[1]) + (a[2] + a[3]);
    __shared__ float red[128];
    red[t] = s;
    __syncthreads();
    for (int o = 64; o > 0; o >>= 1) {
        if (t < o) red[t] += red[t + o];
        __syncthreads();
    }
    if (t == 0) partial[row] = red[0];
}

// ---------------- Kernel 3: deterministic loss reduction ----------------
__global__ void vq_loss_kernel(const float* __restrict__ partial,
                               float* __restrict__ lossOut) {
    const int t = threadIdx.x; // 1024
    float s = 0.f;
    for (int i = t; i < NROWS; i += 1024) s += partial[i];
    __shared__ float red[1024];
    red[t] = s;
    __syncthreads();
    for (int o = 512; o > 0; o >>= 1) {
        if (t < o) red[t] += red[t + o];
        __syncthreads();
    }
    // vq_loss = codebook_loss + 0.25*commitment_loss = 1.25 * mean(diff^2)
    if (t == 0) lossOut[0] = 1.25f * red[0] / 16777216.0f;
}

extern "C" void kernel_launch(void* const* d_in, const int* in_sizes, int n_in,
                              void* d_out, int out_size, void* d_ws, size_t ws_size,
                              hipStream_t stream) {
    (void)in_sizes; (void)n_in; (void)out_size; (void)ws_size;
    const float* ze = (const float*)d_in[0];   // (8,4096,512) f32
    const float* cb = (const float*)d_in[1];   // (8192,512)  f32

    float* out     = (float*)d_out;
    float* zq      = out;                                   // 16,777,216
    float* idxF    = out + (size_t)NROWS * CODE_DIM;        // 32,768
    float* loss    = idxF + NROWS;                          // 1

    float* csq     = (float*)d_ws;                          // 8192 f32
    float* partial = csq + NUM_CODES;                       // 32768 f32

    vq_csq_kernel   <<<NUM_CODES, 128, 0, stream>>>(cb, csq);
    vq_argmin_kernel<<<NROWS / 32, 256, 0, stream>>>(ze, cb, csq, idxF);
    vq_gather_kernel<<<NROWS, 128, 0, stream>>>(ze, cb, idxF, zq, partial);
    vq_loss_kernel  <<<1, 1024, 0, stream>>>(partial, loss);
}